// Net_51307679318827
// MI455X (gfx1250) — compile-verified
//
#include <hip/hip_runtime.h>
#include <hip/hip_bf16.h>

// ---------------- problem constants (match reference) ----------------
#define N_NODES   200000
#define N_EDGES   6400000
#define IN_FEAT   128
#define HIDDEN    16
#define N_CLASSES 32
#define N_TILES   (N_NODES / 16)   // 12500, N_NODES divisible by 16

typedef __attribute__((ext_vector_type(2))) float v2f;
typedef __attribute__((ext_vector_type(8))) float v8f;

// ---------------- init: deg=1 (self loop), zero agg1 and out ----------------
__global__ void init_kernel(float* __restrict__ deg, float* __restrict__ agg1,
                            float* __restrict__ out) {
    unsigned i = blockIdx.x * blockDim.x + threadIdx.x;
    if (i < N_NODES) deg[i] = 1.0f;
    if (i < N_NODES * HIDDEN) agg1[i] = 0.0f;
    if (i < N_NODES * N_CLASSES) out[i] = 0.0f;
}

// ---------------- weighted in-degree: deg[col[e]] += w[e] ----------------
__global__ void deg_kernel(const int* __restrict__ col, const float* __restrict__ w,
                           float* __restrict__ deg) {
    unsigned e = blockIdx.x * blockDim.x + threadIdx.x;
    if (e < N_EDGES) atomicAdd(&deg[col[e]], w[e]);
}

// ---------------- dinv = deg>0 ? rsqrt(deg) : 0 ----------------
__global__ void dinv_kernel(const float* __restrict__ deg, float* __restrict__ dinv) {
    unsigned i = blockIdx.x * blockDim.x + threadIdx.x;
    if (i < N_NODES) {
        float d = deg[i];
        dinv[i] = (d > 0.0f) ? rsqrtf(d) : 0.0f;
    }
}

// ---------------- norm[e] = dinv[row]*w*dinv[col] ----------------
__global__ void norm_kernel(const int* __restrict__ row, const int* __restrict__ col,
                            const float* __restrict__ w, const float* __restrict__ dinv,
                            float* __restrict__ norm) {
    unsigned e = blockIdx.x * blockDim.x + threadIdx.x;
    if (e < N_EDGES) norm[e] = dinv[row[e]] * w[e] * dinv[col[e]];
}

// ---------------- GEMM1: h1[N,16] = x[N,128] @ W1[128,16] via f32 WMMA ----------------
// One wave per 16-row tile; K consumed 4 at a time with V_WMMA_F32_16X16X4_F32.
__global__ void gemm1_wmma(const float* __restrict__ x, const float* __restrict__ W,
                           float* __restrict__ h) {
    unsigned wave = (blockIdx.x * blockDim.x + threadIdx.x) >> 5;
    unsigned lane = threadIdx.x & 31;
    if (wave >= N_TILES) return;                 // wave-uniform exit: EXEC all-1 at WMMA
    unsigned m0 = wave * 16;
    unsigned lh = lane & 15;                     // M index (A) / N index (B,D)
    unsigned hi = lane >> 4;                     // half-wave select
    v8f c = {};
    const float* xrow = x + (m0 + lh) * IN_FEAT; // A: row m0+lh
#pragma unroll
    for (int k = 0; k < IN_FEAT; k += 4) {
        int ka = k + (int)hi * 2;                // lanes16-31 carry K+2,K+3
        v2f a = { xrow[ka], xrow[ka + 1] };
        v2f b = { W[ka * HIDDEN + lh], W[(ka + 1) * HIDDEN + lh] };
        c = __builtin_amdgcn_wmma_f32_16x16x4_f32(false, a, false, b,
                                                  (short)0, c, false, false);
    }
    unsigned mbase = m0 + hi * 8;                // D: vgpr r -> M=r (lo lanes) / r+8 (hi lanes)
#pragma unroll
    for (int r = 0; r < 8; ++r) h[(mbase + r) * HIDDEN + lh] = c[r];
}

// ---------------- scatter1: agg1[col,f] += h1[row,f]*norm, 16 lanes/edge ----------------
__global__ void scatter1_kernel(const int* __restrict__ row, const int* __restrict__ col,
                                const float* __restrict__ norm, const float* __restrict__ h,
                                float* __restrict__ agg) {
    unsigned t = blockIdx.x * blockDim.x + threadIdx.x;
    if (t < (unsigned)N_EDGES * HIDDEN) {
        unsigned e = t >> 4, f = t & 15;
        int r = row[e], c = col[e];
        atomicAdd(&agg[(unsigned)c * HIDDEN + f], h[(unsigned)r * HIDDEN + f] * norm[e]);
    }
}

// ---------------- finalize1 (in place): h2 = relu(agg1 + h1*dinv^2 + b1) ----------------
__global__ void finalize1_kernel(float* __restrict__ agg1, const float* __restrict__ h1,
                                 const float* __restrict__ dinv, const float* __restrict__ b1) {
    unsigned t = blockIdx.x * blockDim.x + threadIdx.x;
    if (t < N_NODES * HIDDEN) {
        unsigned i = t >> 4, f = t & 15;
        float di = dinv[i];
        float v = agg1[t] + h1[t] * di * di + b1[f];
        agg1[t] = v > 0.0f ? v : 0.0f;
    }
}

// ---------------- GEMM2: z[N,32] = h2[N,16] @ W2[16,32] via f32 WMMA ----------------
// One wave per (16-row tile, 16-col tile); 2 col tiles.
__global__ void gemm2_wmma(const float* __restrict__ h2, const float* __restrict__ W,
                           float* __restrict__ z) {
    unsigned gw = (blockIdx.x * blockDim.x + threadIdx.x) >> 5;
    unsigned lane = threadIdx.x & 31;
    if (gw >= N_TILES * 2) return;               // wave-uniform
    unsigned tile = gw >> 1, nt = gw & 1;
    unsigned m0 = tile * 16, n0 = nt * 16;
    unsigned lh = lane & 15, hi = lane >> 4;
    v8f c = {};
    const float* arow = h2 + (m0 + lh) * HIDDEN;
#pragma unroll
    for (int k = 0; k < HIDDEN; k += 4) {
        int ka = k + (int)hi * 2;
        v2f a = { arow[ka], arow[ka + 1] };
        v2f b = { W[ka * N_CLASSES + n0 + lh], W[(ka + 1) * N_CLASSES + n0 + lh] };
        c = __builtin_amdgcn_wmma_f32_16x16x4_f32(false, a, false, b,
                                                  (short)0, c, false, false);
    }
    unsigned mbase = m0 + hi * 8;
#pragma unroll
    for (int r = 0; r < 8; ++r) z[(mbase + r) * N_CLASSES + n0 + lh] = c[r];
}

// ---------------- scatter2: out[col,f] += z[row,f]*norm, 32 lanes/edge ----------------
__global__ void scatter2_kernel(const int* __restrict__ row, const int* __restrict__ col,
                                const float* __restrict__ norm, const float* __restrict__ z,
                                float* __restrict__ out) {
    unsigned t = blockIdx.x * blockDim.x + threadIdx.x;
    if (t < (unsigned)N_EDGES * N_CLASSES) {
        unsigned e = t >> 5, f = t & 31;
        int r = row[e], c = col[e];
        atomicAdd(&out[(unsigned)c * N_CLASSES + f],
                  z[(unsigned)r * N_CLASSES + f] * norm[e]);
    }
}

// ---------------- finalize2: out += z*dinv^2 + b2 ----------------
__global__ void finalize2_kernel(float* __restrict__ out, const float* __restrict__ z,
                                 const float* __restrict__ dinv, const float* __restrict__ b2) {
    unsigned t = blockIdx.x * blockDim.x + threadIdx.x;
    if (t < N_NODES * N_CLASSES) {
        unsigned i = t >> 5, f = t & 31;
        float di = dinv[i];
        out[t] += z[t] * di * di + b2[f];
    }
}

extern "C" void kernel_launch(void* const* d_in, const int* in_sizes, int n_in,
                              void* d_out, int out_size, void* d_ws, size_t ws_size,
                              hipStream_t stream) {
    const float* x  = (const float*)d_in[0];
    const int*   ei = (const int*)d_in[1];          // [2, E] (harness: integer -> int32)
    const float* ew = (const float*)d_in[2];
    const float* W1 = (const float*)d_in[3];
    const float* b1 = (const float*)d_in[4];
    const float* W2 = (const float*)d_in[5];
    const float* b2 = (const float*)d_in[6];
    const int* row = ei;
    const int* col = ei + N_EDGES;
    float* out = (float*)d_out;

    // workspace layout (floats): deg N | dinv N | h1 N*16 | agg1/h2 N*16 | z N*32 | norm E
    float* ws   = (float*)d_ws;
    float* deg  = ws;
    float* dinv = deg  + N_NODES;
    float* h1   = dinv + N_NODES;
    float* agg1 = h1   + (size_t)N_NODES * HIDDEN;  // reused as h2 after finalize1
    float* z    = agg1 + (size_t)N_NODES * HIDDEN;
    float* norm = z    + (size_t)N_NODES * N_CLASSES;

    const int B = 256;
    // init deg / agg1 / out
    init_kernel<<<(N_NODES * N_CLASSES + B - 1) / B, B, 0, stream>>>(deg, agg1, out);
    // weighted degree + normalization
    deg_kernel<<<(N_EDGES + B - 1) / B, B, 0, stream>>>(col, ew, deg);
    dinv_kernel<<<(N_NODES + B - 1) / B, B, 0, stream>>>(deg, dinv);
    norm_kernel<<<(N_EDGES + B - 1) / B, B, 0, stream>>>(row, col, ew, dinv, norm);
    // layer 1
    gemm1_wmma<<<(N_TILES * 32 + B - 1) / B, B, 0, stream>>>(x, W1, h1);
    scatter1_kernel<<<((unsigned)N_EDGES * HIDDEN + B - 1) / B, B, 0, stream>>>(row, col, norm, h1, agg1);
    finalize1_kernel<<<(N_NODES * HIDDEN + B - 1) / B, B, 0, stream>>>(agg1, h1, dinv, b1);
    // layer 2
    gemm2_wmma<<<(N_TILES * 2 * 32 + B - 1) / B, B, 0, stream>>>(agg1, W2, z);
    scatter2_kernel<<<((unsigned)N_EDGES * N_CLASSES + B - 1) / B, B, 0, stream>>>(row, col, norm, z, out);
    finalize2_kernel<<<(N_NODES * N_CLASSES + B - 1) / B, B, 0, stream>>>(out, z, dinv, b2);
}